// SparseMoEBlock_55070070669773
// MI455X (gfx1250) — compile-verified
//
#include <hip/hip_runtime.h>
#include <hip/hip_bf16.h>

// ---------------------------------------------------------------------------
// Sparse MoE (noisy top-2) for MI455X / gfx1250, f32 WMMA path.
//   kernel 1: zero output + expert counters
//   kernel 2: router  (gate logits, noisy top-2, softmax, token lists)
//   kernel 3: fused expert FFN using V_WMMA_F32_16X16X4_F32, X tile staged
//             via TDM gather (tensor_load_to_lds, 6-arg clang-23 form).
// ---------------------------------------------------------------------------

typedef float v2f __attribute__((ext_vector_type(2)));
typedef float v8f __attribute__((ext_vector_type(8)));
typedef unsigned int v4u __attribute__((ext_vector_type(4)));
typedef int v4i __attribute__((ext_vector_type(4)));
typedef int v8i __attribute__((ext_vector_type(8)));

constexpr int T    = 2048;        // B*S tokens
constexpr int D    = 768;
constexpr int E    = 8;
constexpr int DFF  = 3072;
constexpr int MT   = 32;          // tokens per tile
constexpr int FC   = 128;        // DFF chunk size
// X tile row stride in LDS: 768 + 3 pads of 4 dwords (TDM pad: every 256 dw).
// 780 mod 64 = 12 -> 12*t mod 64 distinct for t=0..15 => conflict-free b64 reads.
constexpr int XPR  = 780;
constexpr int HP   = MT + 4;      // padded H row (36: no half-wave conflicts)
constexpr int TILES = T / MT;     // 64 = worst-case tiles per expert

__device__ __forceinline__ v8f wmma4(v2f a, v2f b, v8f c) {
  // D(16x16,f32) = A(16x4,f32) * B(4x16,f32) + C
  return __builtin_amdgcn_wmma_f32_16x16x4_f32(
      /*neg_a=*/false, a, /*neg_b=*/false, b,
      /*c_mod=*/(short)0, c, /*reuse_a=*/false, /*reuse_b=*/false);
}

#if defined(__gfx1250__) && __has_builtin(__builtin_amdgcn_tensor_load_to_lds) && \
    __has_builtin(__builtin_amdgcn_s_wait_tensorcnt)
#define USE_TDM_GATHER 1
#else
#define USE_TDM_GATHER 0
#endif

// ---------------------------------------------------------------------------
__global__ void zero_init_kernel(float* __restrict__ out, int n,
                                 int* __restrict__ counts) {
  int i = blockIdx.x * blockDim.x + threadIdx.x;
  if (i < E) counts[i] = 0;
  for (int j = i; j < n; j += gridDim.x * blockDim.x) out[j] = 0.f;
}

// ---------------------------------------------------------------------------
// Router: one wave (32 lanes) per token.
__global__ __launch_bounds__(256) void router_kernel(
    const float* __restrict__ x, const float* __restrict__ noise,
    const float* __restrict__ Wg, const float* __restrict__ bg,
    const float* __restrict__ Wn, const float* __restrict__ bn,
    float* __restrict__ gates, int* __restrict__ counts,
    int* __restrict__ lists) {
  const int wave = threadIdx.x >> 5;
  const int lane = threadIdx.x & 31;
  const int t = blockIdx.x * 8 + wave;
  if (t >= T) return;

  float ag[E], an[E];
#pragma unroll
  for (int e = 0; e < E; e++) { ag[e] = 0.f; an[e] = 0.f; }

  const float* xr = x + (size_t)t * D;
  for (int d = lane; d < D; d += 32) {
    const float xv = xr[d];
#pragma unroll
    for (int e = 0; e < E; e++) {
      ag[e] += xv * Wg[e * D + d];
      an[e] += xv * Wn[e * D + d];
    }
  }
#pragma unroll
  for (int off = 16; off > 0; off >>= 1) {
#pragma unroll
    for (int e = 0; e < E; e++) {
      ag[e] += __shfl_xor(ag[e], off, 32);
      an[e] += __shfl_xor(an[e], off, 32);
    }
  }

  if (lane == 0) {
    float v1 = -3.0e38f, v2 = -3.0e38f;
    int i1 = 0, i2 = 0;
    float g8[E];
#pragma unroll
    for (int e = 0; e < E; e++) {
      const float lg = ag[e] + bg[e];
      const float ln = an[e] + bn[e];
      const float sp = (ln > 20.f) ? ln : log1pf(expf(ln));  // softplus
      const float nz = lg + noise[t * E + e] * sp;
      g8[e] = 0.f;
      if (nz > v1)      { v2 = v1; i2 = i1; v1 = nz; i1 = e; }
      else if (nz > v2) { v2 = nz; i2 = e; }
    }
    const float e2 = expf(v2 - v1);
    const float inv = 1.f / (1.f + e2);
    g8[i1] = inv;
    g8[i2] = e2 * inv;
#pragma unroll
    for (int e = 0; e < E; e++) gates[t * E + e] = g8[e];
    const int p1 = atomicAdd(&counts[i1], 1); lists[i1 * T + p1] = t;
    const int p2 = atomicAdd(&counts[i2], 1); lists[i2 * T + p2] = t;
  }
}

// ---------------------------------------------------------------------------
// Fused expert FFN: per block = (expert e, tile of 32 tokens).
//   stage0: gather 32 token rows of X into LDS (TDM gather-mode descriptors)
//   stage1: h^T[f,t] = relu(W1_e[f,:] . X[t,:] + b1)    (WMMA, M=f,N=t,K=d)
//   stage2: y^T[d,t] += W2_e[d,:] . h[t,:]              (WMMA, M=d,N=t,K=f)
//   epilogue: out[t,d] += gate[t,e] * (y + b2[d])       (atomicAdd scatter)
__global__ __launch_bounds__(256) void moe_ffn_kernel(
    const float* __restrict__ x,
    const float* __restrict__ W1, const float* __restrict__ b1,
    const float* __restrict__ W2, const float* __restrict__ b2,
    const float* __restrict__ gates, const int* __restrict__ counts,
    const int* __restrict__ lists, float* __restrict__ out) {
  extern __shared__ float smem[];
  float* Xs = smem;                          // [MT][XPR]  (skewed rows)
  float* Hs = smem + MT * XPR;               // [2][FC][HP] double buffered
  int*   toks = (int*)(Hs + 2 * FC * HP);    // [MT]
  float* gws  = (float*)(toks + MT);         // [MT]

  const int e = blockIdx.y;
  const int tile = blockIdx.x;
  const int ntok = counts[e];
  if (tile * MT >= ntok) return;

  const int tid  = threadIdx.x;
  const int lane = tid & 31;
  const int wave = tid >> 5;
  const int lm   = lane & 15;
  const int half = lane >> 4;   // 0: lanes 0-15, 1: lanes 16-31

  int tkv = 0;
  if (tid < MT) {               // == wave 0 (MT == 32)
    const int row = tile * MT + tid;
    tkv = lists[e * T + (row < ntok ? row : 0)];
    toks[tid] = tkv;
    gws[tid]  = (row < ntok) ? gates[tkv * E + e] : 0.f;  // padded rows -> 0
  }

#if USE_TDM_GATHER
  // ---- stage0: two TDM gather-mode loads (16 token rows each), issued by
  // wave 0 only (tensor ops are per-wave, EXEC-independent).
  if (wave == 0) {
    const unsigned long long ga = (unsigned long long)(uintptr_t)x;
    const unsigned int ldsX = (unsigned int)(uintptr_t)Xs;  // low 32b = LDS addr
#pragma unroll
    for (int grp = 0; grp < 2; grp++) {
      int pair[8];
#pragma unroll
      for (int j = 0; j < 8; j++) {   // pack 16 x u16 row indices (token ids)
        const int t0 = __shfl(tkv, grp * 16 + 2 * j, 32);
        const int t1 = __shfl(tkv, grp * 16 + 2 * j + 1, 32);
        pair[j] = (t0 & 0xFFFF) | (t1 << 16);
      }
      // D# group0: count=1, gather_mode=1 (16-bit idx), lds_addr, 57b gaddr, type=2
      const v4u g0 = { 1u | (1u << 31),
                       ldsX + (unsigned)grp * 16u * XPR * 4u,
                       (unsigned)ga,
                       ((unsigned)(ga >> 32) & 0x01FFFFFFu) | (2u << 30) };
      // D# group1: data_size=4B(2), pad_enable, pad_interval=256dw(7),
      // pad_amount=4dw(3); tensor_dim0=768, tile_dim0=768, tile_dim1=16 idx,
      // tensor_dim0_stride=768.
      const v8i g1 = { (int)((2u << 16) | (1u << 20) | (7u << 22) | (3u << 25)),
                       (int)(768u << 16),   // tensor_dim0[15:0] << 16
                       (int)(1u << 16),     // tensor_dim1[15:0] (ignored)
                       (int)(768u << 16),   // tile_dim0 << 16
                       16,                  // tile_dim1 = #indices
                       768,                 // tensor_dim0_stride[31:0]
                       0, 0 };
      const v4i g2 = { pair[0], pair[1], pair[2], pair[3] };
      const v4i g3 = { pair[4], pair[5], pair[6], pair[7] };
      const v8i g4 = { 0, 0, 0, 0, 0, 0, 0, 0 };  // extended group (unused)
      __builtin_amdgcn_tensor_load_to_lds(g0, g1, g2, g3, g4, 0);
    }
    __builtin_amdgcn_s_wait_tensorcnt(0);
  }
  __syncthreads();
#else
  __syncthreads();
  // Fallback: cooperative float4 staging into the same skewed layout.
  for (int i = tid; i < MT * (D / 4); i += blockDim.x) {
    const int r = i / (D / 4);
    const int c = (i % (D / 4)) * 4;
    const float4 v = *(const float4*)(x + (size_t)toks[r] * D + c);
    *(float4*)(Xs + r * XPR + c + 4 * (c >> 8)) = v;
  }
  __syncthreads();
#endif

  const float* W1e = W1 + (size_t)e * DFF * D;
  const float* W2e = W2 + (size_t)e * D * DFF;
  const float* b1e = b1 + (size_t)e * DFF;
  const float* b2e = b2 + (size_t)e * D;

  const v8f zero8 = {0.f, 0.f, 0.f, 0.f, 0.f, 0.f, 0.f, 0.f};
  v8f yacc[12];   // 6 d-tiles x 2 t-tiles of y^T, persistent over DFF chunks
#pragma unroll
  for (int i = 0; i < 12; i++) yacc[i] = zero8;

  int buf = 0;
  for (int fc = 0; fc < DFF; fc += FC) {
    float* Hw = Hs + buf * FC * HP;
    // ---- stage 1: wave w computes h^T rows [fc + w*16, +16) for both t-tiles
    v8f hacc0 = zero8, hacc1 = zero8;
    const float* w1p = W1e + (size_t)(fc + wave * 16 + lm) * D + 2 * half;
    const float* xb  = Xs + lm * XPR + 2 * half;
#pragma unroll
    for (int kc = 0; kc < 3; kc++) {        // 3 chunks of 256 (skewed layout)
      const float* w1k = w1p + kc * 256;
      const float* xk0 = xb + kc * 260;     // 256 cols + 4 pad dwords
      const float* xk1 = xk0 + 16 * XPR;
#pragma unroll 4
      for (int k0 = 0; k0 < 256; k0 += 4) {
        const v2f a  = *(const v2f*)(w1k + k0);   // A: W1 rows (M=f)
        const v2f bA = *(const v2f*)(xk0 + k0);   // B: X^T, t-tile 0
        const v2f bB = *(const v2f*)(xk1 + k0);   // B: X^T, t-tile 1
        hacc0 = wmma4(a, bA, hacc0);
        hacc1 = wmma4(a, bB, hacc1);
      }
    }
    // bias + relu, write h^T chunk to LDS (buffer `buf`)
#pragma unroll
    for (int r = 0; r < 8; r++) {
      const int fl = wave * 16 + r + 8 * half;   // row within chunk
      const float bias = b1e[fc + fl];
      float h0 = hacc0[r] + bias; h0 = h0 > 0.f ? h0 : 0.f;
      float h1 = hacc1[r] + bias; h1 = h1 > 0.f ? h1 : 0.f;
      Hw[fl * HP + lm]      = h0;
      Hw[fl * HP + 16 + lm] = h1;
    }
    __syncthreads();   // single barrier per chunk (Hs is double buffered)

    // ---- stage 2: accumulate y^T with K = this chunk's 128 f-values
#pragma unroll 2
    for (int k0 = 0; k0 < FC; k0 += 4) {
      const int kr = k0 + 2 * half;
      v2f bm0, bm1;
      bm0.x = Hw[kr * HP + lm];       bm0.y = Hw[(kr + 1) * HP + lm];
      bm1.x = Hw[kr * HP + 16 + lm];  bm1.y = Hw[(kr + 1) * HP + 16 + lm];
#pragma unroll
      for (int j = 0; j < 6; j++) {
        const int d0 = (wave + 8 * j) * 16;
        const v2f a = *(const v2f*)(W2e + (size_t)(d0 + lm) * DFF + fc + kr);
        yacc[j * 2 + 0] = wmma4(a, bm0, yacc[j * 2 + 0]);
        yacc[j * 2 + 1] = wmma4(a, bm1, yacc[j * 2 + 1]);
      }
    }
    buf ^= 1;
  }

  // ---- epilogue: gate-weighted scatter with down-proj bias
#pragma unroll
  for (int j = 0; j < 6; j++) {
    const int d0 = (wave + 8 * j) * 16;
#pragma unroll
    for (int n = 0; n < 2; n++) {
      const int trow = n * 16 + lm;
      const float g = gws[trow];
      float* obase = out + (size_t)toks[trow] * D;
#pragma unroll
      for (int r = 0; r < 8; r++) {
        const int d = d0 + r + 8 * half;
        const float val = yacc[j * 2 + n][r] + b2e[d];
        atomicAdd(obase + d, g * val);
      }
    }
  }
}

// ---------------------------------------------------------------------------
extern "C" void kernel_launch(void* const* d_in, const int* in_sizes, int n_in,
                              void* d_out, int out_size, void* d_ws, size_t ws_size,
                              hipStream_t stream) {
  const float* x  = (const float*)d_in[0];
  const float* nz = (const float*)d_in[1];
  const float* Wg = (const float*)d_in[2];
  const float* bg = (const float*)d_in[3];
  const float* Wn = (const float*)d_in[4];
  const float* bn = (const float*)d_in[5];
  const float* W1 = (const float*)d_in[6];
  const float* b1 = (const float*)d_in[7];
  const float* W2 = (const float*)d_in[8];
  const float* b2 = (const float*)d_in[9];
  float* out = (float*)d_out;

  // workspace layout: counts[16] | lists[E*T] | gates[T*E]
  int*   counts = (int*)d_ws;
  int*   lists  = counts + 16;
  float* gates  = (float*)(lists + E * T);

  zero_init_kernel<<<512, 256, 0, stream>>>(out, T * D, counts);
  router_kernel<<<T / 8, 256, 0, stream>>>(x, nz, Wg, bg, Wn, bn,
                                           gates, counts, lists);

  const size_t shbytes = (size_t)(MT * XPR + 2 * FC * HP) * sizeof(float)
                       + MT * sizeof(int) + MT * sizeof(float);
  (void)hipFuncSetAttribute(reinterpret_cast<const void*>(moe_ffn_kernel),
                            hipFuncAttributeMaxDynamicSharedMemorySize,
                            (int)shbytes);
  dim3 grid(TILES, E);
  moe_ffn_kernel<<<grid, 256, shbytes, stream>>>(x, W1, b1, W2, b2,
                                                 gates, counts, lists, out);
}